// RvNN_BU_84911503442503
// MI455X (gfx1250) — compile-verified
//
#include <hip/hip_runtime.h>
#include <hip/hip_bf16.h>

// RvNN bottom-up reference: the scan result is dead (del h_copy). Live math:
//   he0 = h[:,0,:] (32x768) @ W_emb^T (768x128)   -> fp32 WMMA 16x16x4
//   y   = he0 (32x128) @ W_fc^T (128x2) + b_fc    -> tiny dot in LDS
// ~0.5 MB unique bytes, ~6.3 MFLOP => launch-latency bound; fuse into ONE
// kernel, one 512-thread block (16 wave32s), each wave owns one 16x16 C tile.
// Prefetches are hoisted into a branch-free preamble so the K-loop is pure
// load->wmma (the per-iteration guard previously bloated SALU/branches).

typedef __attribute__((ext_vector_type(2))) float v2f;
typedef __attribute__((ext_vector_type(8))) float v8f;

#define B_   32
#define N_   128
#define FI_  768
#define FO_  128

__global__ void __launch_bounds__(512, 1)
rvnn_bu_fused(const float* __restrict__ h,      // [B,N,FI]
              const float* __restrict__ W_emb,  // [FO,FI]
              const float* __restrict__ W_fc,   // [2,FO]
              const float* __restrict__ b_fc,   // [2]
              float* __restrict__ out)          // [B,2]
{
    __shared__ float he0[B_ * FO_];             // 16 KB of 320 KB WGP LDS

    const int lane = threadIdx.x & 31;
    const int wave = threadIdx.x >> 5;          // 0..15
    const int mt   = wave >> 3;                 // 0..1  (M tiles: 32 rows)
    const int nt   = wave & 7;                  // 0..7  (N tiles: 128 cols)
    const int ml   = lane & 15;
    const int half = lane >> 4;                 // 0 or 1

    // A row for this lane: batch b = mt*16 + ml, node 0 => h[b, 0, :]
    const float* __restrict__ arow = h     + (size_t)(mt * 16 + ml) * (N_ * FI_);
    // B "column-source" row: W_emb[n, :] supplies B[k][n] = W_emb[n*FI + k]
    const float* __restrict__ brow = W_emb + (size_t)(nt * 16 + ml) * FI_;

    // Branch-free prefetch preamble: pull both 3 KB rows toward the WGP
    // (each global_prefetch_b8 covers all 32 lanes' addresses).
#pragma unroll
    for (int j = 64; j < FI_; j += 64) {
        __builtin_prefetch(arow + j, 0, 3);
        __builtin_prefetch(brow + j, 0, 3);
    }

    v8f c = {};
#pragma unroll 16
    for (int k0 = 0; k0 < FI_; k0 += 4) {
        // 16x4 f32 A: lanes 0-15 carry K={k0,k0+1}, lanes 16-31 K={k0+2,k0+3}
        v2f a = *(const v2f*)(arow + k0 + 2 * half);
        // 4x16 f32 B: same K split, N = lane&15
        v2f b = *(const v2f*)(brow + k0 + 2 * half);
        // D = A x B + C   (v_wmma_f32_16x16x4_f32, accumulate over K)
        c = __builtin_amdgcn_wmma_f32_16x16x4_f32(
                /*neg_a=*/false, a, /*neg_b=*/false, b,
                /*c_mod=*/(short)0, c, /*reuse_a=*/false, /*reuse_b=*/false);
    }

    // C/D layout: VGPR r -> row = mt*16 + 8*half + r, col = nt*16 + (lane&15)
    const int col = nt * 16 + ml;
#pragma unroll
    for (int r = 0; r < 8; ++r) {
        const int row = mt * 16 + 8 * half + r;
        he0[row * FO_ + col] = c[r];
    }
    __syncthreads();

    // Head: y[b,k] = b_fc[k] + sum_o he0[b,o] * W_fc[k,o]   (64 outputs)
    if (threadIdx.x < B_ * 2) {
        const int b = threadIdx.x >> 1;
        const int k = threadIdx.x & 1;
        float acc = b_fc[k];
        const float* __restrict__ wrow = W_fc + k * FO_;
        const float* __restrict__ hrow = he0 + b * FO_;
#pragma unroll 8
        for (int o = 0; o < FO_; ++o)
            acc = __builtin_fmaf(hrow[o], wrow[o], acc);
        out[b * 2 + k] = acc;
    }
}

extern "C" void kernel_launch(void* const* d_in, const int* in_sizes, int n_in,
                              void* d_out, int out_size, void* d_ws, size_t ws_size,
                              hipStream_t stream) {
    // setup_inputs order: 0:h 1:adj 2:W_emb 3:W_ioux 4:b_ioux 5:W_iouh 6:b_iouh
    //                     7:W_coux 8:b_coux 9:W_couh 10:b_couh 11:W_fc 12:b_fc
    const float* h     = (const float*)d_in[0];
    const float* W_emb = (const float*)d_in[2];
    const float* W_fc  = (const float*)d_in[11];
    const float* b_fc  = (const float*)d_in[12];
    float*       out   = (float*)d_out;            // [32,2] fp32

    (void)in_sizes; (void)n_in; (void)out_size; (void)d_ws; (void)ws_size;

    rvnn_bu_fused<<<1, 512, 0, stream>>>(h, W_emb, W_fc, b_fc, out);
}